// Mirror_tu_6743098655506
// MI455X (gfx1250) — compile-verified
//
#include <hip/hip_runtime.h>

// Mirror-flip + fp32 -> fp16 cast, per-batch flag controlled.
//
// x        : [16, 2, 64, 128, 128] f32
// x_flag   : [16] f32
// y_flag   : [16] f32
// out[b]   : zeros                      if x_flag[b] <= 0.5
//            flip_h(x[b])               if x_flag[b] >  0.5 && y_flag[b] <= 0.5
//            flip_h(flip_w(x[b]))       if x_flag[b] >  0.5 && y_flag[b] >  0.5
// cast to f16.
//
// Pure streaming kernel: 128 MB read (only for flipped batches) + 64 MB write.
// HBM-bound; use b128 loads/stores with non-temporal hints, block-uniform
// flag branches, wave32-contiguous addressing.

typedef float    __attribute__((ext_vector_type(4))) v4f;
typedef _Float16 __attribute__((ext_vector_type(8))) v8h;

namespace {
constexpr int kB = 16;
constexpr int kC = 2;
constexpr int kD = 64;
constexpr int kW = 128;
constexpr int kH = 128;
constexpr int kElemPerBatch    = kC * kD * kW * kH;      // 2,097,152
constexpr int kBlock           = 256;
constexpr int kThreadsPerBatch = kElemPerBatch / 8;      // 262,144 (8 h-elems/thread)
constexpr int kBlocksPerBatch  = kThreadsPerBatch / kBlock; // 1,024
} // namespace

__global__ __launch_bounds__(kBlock) void mirror_flip_cast_kernel(
    const float* __restrict__ x,
    const float* __restrict__ x_flag,
    const float* __restrict__ y_flag,
    _Float16* __restrict__ out)
{
    const int b = (int)(blockIdx.x >> 10);                       // / kBlocksPerBatch
    const int t = (int)(((blockIdx.x & 1023u) << 8) | threadIdx.x);

    const int h8 = t & 15;          // which group of 8 along h
    const int w  = (t >> 4) & 127;  // 0..127
    const int cd = t >> 11;         // 0..127  (c*kD + d)

    const long base_out = (long)b * kElemPerBatch + (long)cd * (kW * kH);
    v8h* const outp = (v8h*)(out + base_out + (long)w * kH + h8 * 8);

    // Block-uniform flag loads (b is uniform per block -> scalar path).
    const bool do_flip = x_flag[b] > 0.5f;

    if (!do_flip) {
        v8h z = {};
        __builtin_nontemporal_store(z, outp);
        return;
    }

    const bool flip_w = y_flag[b] > 0.5f;
    const int  sw     = flip_w ? (kW - 1 - w) : w;
    const int  hs     = kH - 8 - h8 * 8;     // source start along h (reversed)

    const float* src = x + (long)b * kElemPerBatch + (long)cd * (kW * kH)
                         + (long)sw * kH + hs;

    // Two 128-bit NT loads cover x[hs .. hs+7]; output h-index j maps to
    // source hs + 7 - j, i.e. reverse the 8 elements while converting.
    v4f lo = __builtin_nontemporal_load((v4f*)src);        // x[hs+0 .. hs+3]
    v4f hi = __builtin_nontemporal_load((v4f*)(src + 4));  // x[hs+4 .. hs+7]

    v8h r;
    r[0] = (_Float16)hi[3];
    r[1] = (_Float16)hi[2];
    r[2] = (_Float16)hi[1];
    r[3] = (_Float16)hi[0];
    r[4] = (_Float16)lo[3];
    r[5] = (_Float16)lo[2];
    r[6] = (_Float16)lo[1];
    r[7] = (_Float16)lo[0];

    __builtin_nontemporal_store(r, outp);
}

extern "C" void kernel_launch(void* const* d_in, const int* in_sizes, int n_in,
                              void* d_out, int out_size, void* d_ws, size_t ws_size,
                              hipStream_t stream) {
    (void)in_sizes; (void)n_in; (void)out_size; (void)d_ws; (void)ws_size;

    const float* x      = (const float*)d_in[0];
    const float* x_flag = (const float*)d_in[1];
    const float* y_flag = (const float*)d_in[2];
    _Float16*    out    = (_Float16*)d_out;

    dim3 grid(kB * kBlocksPerBatch);   // 16,384 blocks
    dim3 block(kBlock);                // 256 threads = 8 wave32
    mirror_flip_cast_kernel<<<grid, block, 0, stream>>>(x, x_flag, y_flag, out);
}